// Shake_53377853554895
// MI455X (gfx1250) — compile-verified
//
#include <hip/hip_runtime.h>

// SHAKE constraint solver for MI455X (gfx1250).
// B=8, N=2048, C=512, 3 Newton iterations.
//  - Precompute GT[b][k][c][3] = J[b,c,i_k]/m_i - J[b,c,j_k]/m_j (constant over iters).
//  - Per iteration: GEMV for dif (wave-per-row, shfl reduce), jac via dot, then
//    blocked LU (16x16 blocks, augmented RHS column) with V_WMMA_F32_16X16X4_F32
//    trailing updates fed from LDS panels staged by the Tensor Data Mover
//    (tensor_load_to_lds + s_wait_tensorcnt), one workgroup of 32 waves per batch.
//  - Final streaming pass over jac_prv (100MB, fits in 192MB L2) for pos/mom update.

#define BB 8
#define NN 2048
#define CC 512
#define NITERS 3
#define LDA 544   // 512 matrix cols + col 512 = RHS + pad; multiple of 16

typedef float v2f __attribute__((ext_vector_type(2)));
typedef float v8f __attribute__((ext_vector_type(8)));
typedef unsigned int u32x4 __attribute__((ext_vector_type(4)));
typedef int i32x8 __attribute__((ext_vector_type(8)));
typedef int i32x4 __attribute__((ext_vector_type(4)));

// workspace layout (float offsets)
#define GT_OFF   0
#define GT_SZ    (BB*CC*CC*3)
#define AUG_OFF  (GT_OFF + GT_SZ)
#define AUG_SZ   (BB*CC*LDA)
#define DIF0_OFF (AUG_OFF + AUG_SZ)
#define DIF0_SZ  (BB*CC*3)
#define DIF_OFF  (DIF0_OFF + DIF0_SZ)
#define DIF_SZ   (BB*CC*3)
#define LMD_OFF  (DIF_OFF + DIF_SZ)

#if __has_builtin(__builtin_amdgcn_tensor_load_to_lds)
#define HAVE_TDM 1
#else
#define HAVE_TDM 0
#endif

// ---------------- K0: precompute GT, diff0, zero lambda ----------------
__global__ void k0_prep(const float* __restrict__ pos, const float* __restrict__ mas,
                        const float* __restrict__ jacp, const int* __restrict__ pi,
                        const int* __restrict__ pj, float* __restrict__ ws) {
  int blk = blockIdx.x;            // b*CC + k
  int b = blk / CC, k = blk % CC;
  int i = pi[k], j = pj[k];
  float rmi = 1.0f / mas[b*NN + i];
  float rmj = 1.0f / mas[b*NN + j];
  if (threadIdx.x == 0) {
    float* d0p = ws + DIF0_OFF + (b*CC + k)*3;
    const float* pb = pos + (size_t)b*NN*3;
    d0p[0] = pb[i*3+0] - pb[j*3+0];
    d0p[1] = pb[i*3+1] - pb[j*3+1];
    d0p[2] = pb[i*3+2] - pb[j*3+2];
    ws[LMD_OFF + b*CC + k] = 0.0f;
  }
  const float* Jb = jacp + (size_t)b*CC*NN*3;
  float* GT = ws + GT_OFF;
  for (int c = threadIdx.x; c < CC; c += blockDim.x) {
    const float* Ji = Jb + (size_t)c*NN*3 + (size_t)i*3;
    const float* Jj = Jb + (size_t)c*NN*3 + (size_t)j*3;
    float* g = GT + ((size_t)(b*CC + k)*CC + c)*3;
    g[0] = Ji[0]*rmi - Jj[0]*rmj;
    g[1] = Ji[1]*rmi - Jj[1]*rmj;
    g[2] = Ji[2]*rmi - Jj[2]*rmj;
  }
}

// ---------------- K1: dif + constraint value (RHS) ----------------
__global__ void k1_con(const float* __restrict__ dtm, const float* __restrict__ d0,
                       float* __restrict__ ws) {
  int g = blockIdx.x * (blockDim.x >> 5) + (threadIdx.x >> 5);
  int lane = threadIdx.x & 31;
  int b = g / CC, k = g % CC;
  const float* GT = ws + GT_OFF + (size_t)(b*CC + k)*CC*3;
  const float* lmd = ws + LMD_OFF + b*CC;
  float s0 = 0.f, s1 = 0.f, s2 = 0.f;
  for (int c = lane; c < CC; c += 32) {
    float w = lmd[c];
    const float* gp = GT + c*3;
    s0 += w*gp[0]; s1 += w*gp[1]; s2 += w*gp[2];
  }
  for (int off = 16; off; off >>= 1) {
    s0 += __shfl_xor(s0, off, 32);
    s1 += __shfl_xor(s1, off, 32);
    s2 += __shfl_xor(s2, off, 32);
  }
  if (lane == 0) {
    float dt = dtm[b], dt2 = dt*dt;
    const float* df0 = ws + DIF0_OFF + (b*CC + k)*3;
    float e0 = df0[0] - dt2*s0;
    float e1 = df0[1] - dt2*s1;
    float e2 = df0[2] - dt2*s2;
    float* dif = ws + DIF_OFF + (b*CC + k)*3;
    dif[0] = e0; dif[1] = e1; dif[2] = e2;
    ws[AUG_OFF + (size_t)b*CC*LDA + (size_t)k*LDA + 512] =
        e0*e0 + e1*e1 + e2*e2 - d0[k];
  }
}

// ---------------- K2: Jacobian row (and zero pad cols) ----------------
__global__ void k2_jac(const float* __restrict__ dtm, float* __restrict__ ws) {
  int blk = blockIdx.x;
  int b = blk / CC, k = blk % CC;
  int c = threadIdx.x;               // 0..543
  float* Arow = ws + AUG_OFF + (size_t)b*CC*LDA + (size_t)k*LDA;
  if (c > 512) { Arow[c] = 0.0f; return; }   // pad cols of RHS block
  if (c == 512) return;                      // RHS written by k1
  float dt = dtm[b];
  float m2 = -2.0f*dt*dt;
  const float* dif = ws + DIF_OFF + (b*CC + k)*3;
  const float* g = ws + GT_OFF + ((size_t)(b*CC + k)*CC + c)*3;
  Arow[c] = m2 * (dif[0]*g[0] + dif[1]*g[1] + dif[2]*g[2]);
}

#if HAVE_TDM
// Issue one TDM 2D tile load: tile1 rows of tile0 f32 elements, row stride
// stride_elems, global -> LDS (packed rows). Wave-level op; args are uniform.
__device__ inline void tdm_load_2d(const float* gbase, unsigned lds_off,
                                   int tile0, int tile1, int stride_elems) {
  unsigned long long ga = (unsigned long long)(size_t)gbase;
  u32x4 g0;
  g0.x = 1u;                                                // count=1, user mode
  g0.y = lds_off;                                           // lds_addr (bytes)
  g0.z = (unsigned)(ga & 0xFFFFFFFFu);                      // global_addr lo
  g0.w = (unsigned)((ga >> 32) & 0x1FFFFFFu) | (2u << 30);  // addr hi | type=2
  unsigned td0 = (unsigned)tile0, td1 = (unsigned)tile1;
  i32x8 g1;
  g1[0] = (int)(2u << 16);                                  // data_size=4B, mask=0
  g1[1] = (int)((td0 & 0xFFFFu) << 16);                     // tensor_dim0 lo16
  g1[2] = (int)(((td0 >> 16) & 0xFFFFu) | ((td1 & 0xFFFFu) << 16));
  g1[3] = (int)(((td1 >> 16) & 0xFFFFu) | ((td0 & 0xFFFFu) << 16)); // tile_dim0
  g1[4] = (int)(td1 & 0xFFFFu);                             // tile_dim1, tile_dim2=0
  g1[5] = stride_elems;                                     // tensor_dim0_stride lo32
  g1[6] = 0;
  g1[7] = 0;
  i32x4 z4 = {0, 0, 0, 0};
  i32x8 z8 = {0, 0, 0, 0, 0, 0, 0, 0};
  // 6-arg form (clang-23 / therock-10.0 headers): g0, g1, g2, g3, extra, cpol
  __builtin_amdgcn_tensor_load_to_lds(g0, g1, z4, z4, z8, 0);
}
#endif

__device__ inline void wait_tensor0() {
#if __has_builtin(__builtin_amdgcn_s_wait_tensorcnt)
  __builtin_amdgcn_s_wait_tensorcnt(0);
#else
  asm volatile("s_wait_tensorcnt 0x0" ::: "memory");
#endif
}

// ---------------- K3: blocked LU solve, WMMA trailing updates from LDS ----------------
__global__ void __launch_bounds__(1024)
k3_solve(float* __restrict__ ws) {
  __shared__ float sB[16 * 512];   // U12 panel: 16 rows x (ncb*16) cols, packed
  __shared__ float sA[496 * 16];   // L21 panel: (nrb*16) rows x 16 cols, packed
  int b = blockIdx.x;
  float* A = ws + AUG_OFF + (size_t)b*CC*LDA;
  int tid = threadIdx.x;
  int wave = tid >> 5, lane = tid & 31;
  int sub = lane >> 4, l16 = lane & 15;
#define AG(r, c) A[(size_t)(r)*LDA + (c)]

  for (int kb = 0; kb < 32; ++kb) {
    int k0 = kb*16;
    int ncb = 32 - kb;             // col blocks kb+1..32 (incl. RHS block)
    int nrb = 31 - kb;             // row blocks kb+1..31
    int w = ncb*16;

    // step 1: factor 16x16 diagonal block (wave 0, lanes 0..15)
    if (wave == 0) {
      for (int i = 0; i < 16; ++i) {
        float inv = 1.0f / AG(k0+i, k0+i);
        if (lane > i && lane < 16) {
          float l = AG(k0+lane, k0+i) * inv;
          AG(k0+lane, k0+i) = l;
          for (int c2 = i+1; c2 < 16; ++c2)
            AG(k0+lane, k0+c2) -= l * AG(k0+i, k0+c2);
        }
      }
    }
    __threadfence_block();
    __syncthreads();

    // step 2: row panel U12 (col blocks kb+1..32): X := L11^-1 X
    {
      int q = wave*2 + sub;
      if (q < ncb) {
        int col = (kb+1+q)*16 + l16;
        float x[16];
        #pragma unroll
        for (int r = 0; r < 16; ++r) x[r] = AG(k0+r, col);
        for (int i = 0; i < 16; ++i)
          for (int r = i+1; r < 16; ++r)
            x[r] -= AG(k0+r, k0+i) * x[i];
        #pragma unroll
        for (int r = 0; r < 16; ++r) AG(k0+r, col) = x[r];
      }
    }
    // step 3: column panel L21 (row blocks kb+1..31): X := X U11^-1
    {
      int q = wave*2 + sub;
      if (q < nrb) {
        int row = (kb+1+q)*16 + l16;
        float x[16];
        #pragma unroll
        for (int jc = 0; jc < 16; ++jc) x[jc] = AG(row, k0+jc);
        for (int jc = 0; jc < 16; ++jc) {
          float s = x[jc];
          for (int i2 = 0; i2 < jc; ++i2) s -= x[i2] * AG(k0+i2, k0+jc);
          x[jc] = s / AG(k0+jc, k0+jc);
        }
        #pragma unroll
        for (int jc = 0; jc < 16; ++jc) AG(row, k0+jc) = x[jc];
      }
    }
    __threadfence_block();
    __syncthreads();

    // stage panels into LDS for the trailing update
    if (nrb > 0) {
#if HAVE_TDM
      if (wave == 0) {
        // U12: 16 rows x w cols starting at (k0, k0+16)
        tdm_load_2d(&AG(k0, k0 + 16), (unsigned)(size_t)sB, w, 16, LDA);
        // L21: nrb*16 rows x 16 cols starting at (k0+16, k0)
        tdm_load_2d(&AG(k0 + 16, k0), (unsigned)(size_t)sA, 16, nrb*16, LDA);
        wait_tensor0();
      }
#else
      for (int idx = tid; idx < 16*w; idx += 1024) {
        int r = idx / w, c2 = idx % w;
        sB[idx] = AG(k0 + r, k0 + 16 + c2);
      }
      for (int idx = tid; idx < nrb*16*16; idx += 1024) {
        int r = idx >> 4, c2 = idx & 15;
        sA[idx] = AG(k0 + 16 + r, k0 + c2);
      }
#endif
    }
    __syncthreads();

    // step 4: trailing update A22 -= L21*U12 via chained V_WMMA_F32_16X16X4_F32,
    // A/B operands from LDS, C tile from global. Per-wave tiles => EXEC all-1s.
    {
      int ntiles = nrb * ncb;
      int m = l16, n = l16, half = sub;
      for (int t = wave; t < ntiles; t += 32) {
        int ia = t / ncb;          // 0..nrb-1
        int ja = t % ncb;          // 0..ncb-1
        int r0 = (kb + 1 + ia)*16, c0 = (kb + 1 + ja)*16;
        v8f acc;
        #pragma unroll
        for (int r = 0; r < 8; ++r) acc[r] = -AG(r0 + r + 8*half, c0 + n);
        #pragma unroll
        for (int kk = 0; kk < 4; ++kk) {
          int kc = kk*4 + 2*half;
          v2f av, bv;
          av.x = sA[(ia*16 + m)*16 + kc];
          av.y = sA[(ia*16 + m)*16 + kc + 1];
          bv.x = sB[(kc    )*w + ja*16 + n];
          bv.y = sB[(kc + 1)*w + ja*16 + n];
          acc = __builtin_amdgcn_wmma_f32_16x16x4_f32(false, av, false, bv,
                                                      (short)0, acc, false, false);
        }
        #pragma unroll
        for (int r = 0; r < 8; ++r) AG(r0 + r + 8*half, c0 + n) = -acc[r];
      }
    }
    __threadfence_block();
    __syncthreads();
  }

  // back substitution: col 512 holds y = L^-1 b; solve U x = y in place
  for (int k = CC-1; k >= 0; --k) {
    if (tid == 0) AG(k, 512) = AG(k, 512) / AG(k, k);
    __threadfence_block();
    __syncthreads();
    float xk = AG(k, 512);
    if (tid < k) AG(tid, 512) -= AG(tid, k) * xk;
    __threadfence_block();
    __syncthreads();
  }
  if (tid < CC) ws[LMD_OFF + b*CC + tid] -= AG(tid, 512);
#undef AG
}

// ---------------- K4: final pos/mom update (streams jac_prv, L2-resident) ----------------
__global__ void k4_final(const float* __restrict__ pos, const float* __restrict__ mom,
                         const float* __restrict__ mas, const float* __restrict__ dtm,
                         const float* __restrict__ jacp, const float* __restrict__ ws,
                         float* __restrict__ out) {
  int b = blockIdx.y;
  int t = blockIdx.x*blockDim.x + threadIdx.x;    // 0..N*3-1
  if (t >= NN*3) return;
  int n = t / 3;
  const float* lmd = ws + LMD_OFF + b*CC;
  const float* Jb = jacp + (size_t)b*CC*NN*3 + t;
  float acc = 0.0f;
  #pragma unroll 8
  for (int c = 0; c < CC; ++c)
    acc += lmd[c] * Jb[(size_t)c*NN*3];
  float frc = -acc;
  float dt = dtm[b], dt2 = dt*dt;
  float rm = 1.0f / mas[b*NN + n];
  size_t gi = (size_t)b*NN*3 + t;
  out[gi] = pos[gi] + frc*rm*dt2;
  out[(size_t)BB*NN*3 + gi] = mom[gi] + frc*dt;
}

// ---------------- K5: emit lambda ----------------
__global__ void k5_lmd(const float* __restrict__ ws, float* __restrict__ out) {
  int t = blockIdx.x*blockDim.x + threadIdx.x;
  if (t < BB*CC) out[(size_t)2*BB*NN*3 + t] = ws[LMD_OFF + t];
}

extern "C" void kernel_launch(void* const* d_in, const int* in_sizes, int n_in,
                              void* d_out, int out_size, void* d_ws, size_t ws_size,
                              hipStream_t stream) {
  const float* pos  = (const float*)d_in[0];
  const float* mom  = (const float*)d_in[1];
  const float* mas  = (const float*)d_in[2];
  const float* dtm  = (const float*)d_in[3];
  const float* jacp = (const float*)d_in[4];
  const int*   pi   = (const int*)d_in[5];
  const int*   pj   = (const int*)d_in[6];
  const float* d0   = (const float*)d_in[7];
  float* out = (float*)d_out;
  float* ws  = (float*)d_ws;

  k0_prep<<<dim3(BB*CC), dim3(256), 0, stream>>>(pos, mas, jacp, pi, pj, ws);
  for (int it = 0; it < NITERS; ++it) {
    k1_con<<<dim3(BB*CC/8), dim3(256), 0, stream>>>(dtm, d0, ws);
    k2_jac<<<dim3(BB*CC), dim3(544), 0, stream>>>(dtm, ws);
    k3_solve<<<dim3(BB), dim3(1024), 0, stream>>>(ws);
  }
  k4_final<<<dim3((NN*3 + 255)/256, BB), dim3(256), 0, stream>>>(pos, mom, mas, dtm, jacp, ws, out);
  k5_lmd<<<dim3((BB*CC + 255)/256), dim3(256), 0, stream>>>(ws, out);
}